// BatchedSemiAttention_54099408060543
// MI455X (gfx1250) — compile-verified
//
#include <hip/hip_runtime.h>
#include <math.h>

#define NSEG 128
#define DDIM 256
#define EDIM 256
#define R3   1024        // rows per block in the main accumulation pass
#define TROWS 32         // rows per async stage tile
#define NTILES (R3 / TROWS)
#define ASY_PER_TILE ((TROWS * EDIM * 4) / (256 * 16))   // 8 b128 asyncs/thread/tile

#if defined(__has_builtin)
#if __has_builtin(__builtin_amdgcn_global_load_async_to_lds_b128) && \
    __has_builtin(__builtin_amdgcn_s_wait_asynccnt)
#define HAVE_ASYNC 1
#endif
#endif

#define AS1 __attribute__((address_space(1)))
#define AS3 __attribute__((address_space(3)))

typedef int b128_t __attribute__((__vector_size__(16)));   // matches builtin param
typedef __attribute__((ext_vector_type(2))) float v2f;
typedef __attribute__((ext_vector_type(8))) float v8f;

// ---- helpers -------------------------------------------------------------

__device__ __forceinline__ float waveReduceAdd(float v) {
#pragma unroll
  for (int off = 16; off > 0; off >>= 1) v += __shfl_xor(v, off, 32);
  return v;
}

// float atomic max via sign-split int trick (works with init = -inf)
__device__ __forceinline__ void atomicMaxF(float* addr, float val) {
  if (val >= 0.0f)
    atomicMax((int*)addr, __float_as_int(val));
  else
    atomicMin((unsigned int*)addr, __float_as_uint(val));
}

__device__ __forceinline__ void atomAddF(float* p, float v) {
  unsafeAtomicAdd(p, v);   // native global_atomic_add_f32 on gfx1250
}

// ---- kernel 0: prep (weight folds + re-init of accumulators) -------------

__global__ void k_prep(const float* __restrict__ Wk, const float* __restrict__ bk,
                       float* __restrict__ wksum, float* __restrict__ bksum,
                       float* __restrict__ segmax, float* __restrict__ segsum,
                       float* __restrict__ A) {
  int t = threadIdx.x;
  float s = 0.f;
  const float* row = Wk + (size_t)t * EDIM;
#pragma unroll 8
  for (int e = 0; e < EDIM; ++e) s += row[e];
  wksum[t] = s;

  __shared__ float red[256];
  red[t] = bk[t];
  __syncthreads();
  for (int off = 128; off > 0; off >>= 1) {
    if (t < off) red[t] += red[t + off];
    __syncthreads();
  }
  if (t == 0) bksum[0] = red[0];

  if (t < NSEG) { segmax[t] = -INFINITY; segsum[t] = 0.f; }
  for (int i = t; i < NSEG * EDIM; i += 256) A[i] = 0.f;
}

// ---- kernel 1: logits = x @ wksum + bksum ; per-segment max --------------

__global__ void k_logits(const float* __restrict__ x, const int* __restrict__ seg,
                         const float* __restrict__ wksum, const float* __restrict__ bksum,
                         float* __restrict__ logits, float* __restrict__ segmax, int N) {
  __shared__ __align__(16) float wk[DDIM];
  int t = threadIdx.x;
  wk[t] = wksum[t];
  __syncthreads();

  int wave = t >> 5, lane = t & 31;
  int base = blockIdx.x * 64 + wave * 8;
  float bks = bksum[0];
  const float4* wk4 = (const float4*)wk;
  float4 w0 = wk4[lane * 2];
  float4 w1 = wk4[lane * 2 + 1];

  int curSeg = -1; float curMax = -INFINITY;
  for (int j = 0; j < 8; ++j) {
    int rowi = base + j;
    if (rowi >= N) break;
    const float4* xr = (const float4*)(x + (size_t)rowi * DDIM);
    float4 a0 = xr[lane * 2];
    float4 a1 = xr[lane * 2 + 1];
    float p = a0.x * w0.x + a0.y * w0.y + a0.z * w0.z + a0.w * w0.w
            + a1.x * w1.x + a1.y * w1.y + a1.z * w1.z + a1.w * w1.w;
    p = waveReduceAdd(p);
    float lg = p + bks;
    int s = seg[rowi];
    if (lane == 0) logits[rowi] = lg;
    if (s != curSeg) {
      if (curSeg >= 0 && lane == 0) atomicMaxF(&segmax[curSeg], curMax);
      curSeg = s; curMax = lg;
    } else {
      curMax = fmaxf(curMax, lg);
    }
  }
  if (curSeg >= 0 && lane == 0) atomicMaxF(&segmax[curSeg], curMax);
}

// ---- kernel 2: segsum[s] = sum exp(logit - segmax[s]) --------------------

__global__ void k_segsum(const float* __restrict__ logits, const int* __restrict__ seg,
                         const float* __restrict__ segmax, float* __restrict__ segsum,
                         int N) {
  __shared__ float ls[NSEG];
  int t = threadIdx.x;
  if (t < NSEG) ls[t] = 0.f;
  __syncthreads();

  int base = blockIdx.x * 4096;
  int cur = -1; float acc = 0.f;
#pragma unroll 4
  for (int j = 0; j < 16; ++j) {
    int i = base + t + j * 256;
    if (i >= N) break;
    int s = seg[i];
    float e = __expf(logits[i] - segmax[s]);
    if (s != cur) {
      if (cur >= 0) atomicAdd(&ls[cur], acc);
      cur = s; acc = e;
    } else {
      acc += e;
    }
  }
  if (cur >= 0) atomicAdd(&ls[cur], acc);
  __syncthreads();
  if (t < NSEG && ls[t] != 0.f) atomAddF(&segsum[t], ls[t]);
}

// ---- kernel 3: A[s,:] += attn_i * x[i,:]  (main bandwidth pass) ----------
// Async-to-LDS double-buffered staging when the gfx1250 builtins are
// available; direct float4 streaming fallback otherwise.

__global__ void k_accum(const float* __restrict__ x, const int* __restrict__ seg,
                        const float* __restrict__ logits, const float* __restrict__ segmax,
                        const float* __restrict__ segsum, float* __restrict__ A, int N) {
  __shared__ float wbuf[R3];
  __shared__ int   sbuf[R3];
  int t = threadIdx.x;
  int r0 = blockIdx.x * R3;

  // Phase A: per-row softmax weights into LDS (cheap; reads logits/seg only)
#pragma unroll
  for (int j = 0; j < R3 / 256; ++j) {
    int r = t + j * 256;
    int i = r0 + r;
    if (i < N) {
      int s = seg[i];
      wbuf[r] = __expf(logits[i] - segmax[s]) / segsum[s];
      sbuf[r] = s;
      __builtin_prefetch(x + (size_t)i * EDIM, 0, 0);  // global_prefetch_b8
    } else {
      wbuf[r] = 0.f; sbuf[r] = NSEG - 1;
    }
  }
  __syncthreads();

#if defined(HAVE_ASYNC)
  if (r0 + R3 <= N) {
    extern __shared__ float stage[];   // 2 * TROWS * EDIM floats (64 KB dynamic)
    const char* gbase = (const char*)x + (size_t)r0 * EDIM * 4 + (size_t)t * 16;
    char* lbase = (char*)stage + (size_t)t * 16;

    auto issue = [&](int tile) {
      const char* g = gbase + (size_t)tile * TROWS * EDIM * 4;
      char* l = lbase + (size_t)(tile & 1) * TROWS * EDIM * 4;
#pragma unroll
      for (int j = 0; j < ASY_PER_TILE; ++j)
        __builtin_amdgcn_global_load_async_to_lds_b128(
            (AS1 b128_t*)(g + j * 4096), (AS3 b128_t*)(l + j * 4096), 0, 0);
    };

    issue(0);
    int d = t;
    float acc = 0.f;
    int cur = sbuf[0];
    for (int tile = 0; tile < NTILES; ++tile) {
      if (tile + 1 < NTILES) {
        issue(tile + 1);
        __builtin_amdgcn_s_wait_asynccnt(ASY_PER_TILE);  // tile's asyncs done
      } else {
        __builtin_amdgcn_s_wait_asynccnt(0);
      }
      __syncthreads();                                   // all threads' asyncs visible
      const float* sb = stage + (size_t)(tile & 1) * (TROWS * EDIM);
      int rb = tile * TROWS;
#pragma unroll 8
      for (int rl = 0; rl < TROWS; ++rl) {
        int r = rb + rl;
        int s = sbuf[r];                  // uniform across block
        if (s != cur) {                   // rare: sorted segment boundary
          atomAddF(&A[(size_t)cur * EDIM + d], acc);
          acc = 0.f; cur = s;
        }
        acc = fmaf(wbuf[r], sb[(size_t)rl * EDIM + d], acc);
      }
      __syncthreads();                    // buffer consumed before next overwrite
    }
    atomAddF(&A[(size_t)cur * EDIM + d], acc);
    return;
  }
#endif
  {
    // Direct-load fallback: 4 row-streams x 64 column-quads, float4 per step
    int h = t >> 6;          // row stream 0..3 (uniform per wave)
    int q = t & 63;          // column quad
    int d4 = q * 4;
    float4 acc4 = {0.f, 0.f, 0.f, 0.f};
    int cur = sbuf[h];
    const float4* xp = (const float4*)(x + (size_t)r0 * EDIM) + q;
    for (int r = h; r < R3; r += 4) {
      if (r0 + r >= N) break;
      int s = sbuf[r];
      if (s != cur) {
        atomAddF(&A[(size_t)cur * EDIM + d4 + 0], acc4.x);
        atomAddF(&A[(size_t)cur * EDIM + d4 + 1], acc4.y);
        atomAddF(&A[(size_t)cur * EDIM + d4 + 2], acc4.z);
        atomAddF(&A[(size_t)cur * EDIM + d4 + 3], acc4.w);
        acc4 = {0.f, 0.f, 0.f, 0.f};
        cur = s;
      }
      float4 v = xp[(size_t)r * (EDIM / 4)];
      float w = wbuf[r];
      acc4.x = fmaf(w, v.x, acc4.x);
      acc4.y = fmaf(w, v.y, acc4.y);
      acc4.z = fmaf(w, v.z, acc4.z);
      acc4.w = fmaf(w, v.w, acc4.w);
    }
    atomAddF(&A[(size_t)cur * EDIM + d4 + 0], acc4.x);
    atomAddF(&A[(size_t)cur * EDIM + d4 + 1], acc4.y);
    atomAddF(&A[(size_t)cur * EDIM + d4 + 2], acc4.z);
    atomAddF(&A[(size_t)cur * EDIM + d4 + 3], acc4.w);
  }
}

// ---- kernel 4: pooled = A @ Wv + bv*ind  via v_wmma_f32_16x16x4_f32 ------

__global__ void k_wmma_pooled(const float* __restrict__ A, const float* __restrict__ Wv,
                              const float* __restrict__ bv, const float* __restrict__ segsum,
                              float* __restrict__ pooled) {
  int wave = threadIdx.x >> 5;
  int lane = threadIdx.x & 31;
  int g = blockIdx.x * (blockDim.x >> 5) + wave;  // 0..127 : one 16x16 tile per wave
  int m0 = (g >> 4) * 16;                         // 8 tile rows  (128 segs)
  int n0 = (g & 15) * 16;                         // 16 tile cols (256 feats)
  int half = lane >> 4;                           // lane halves carry K and K+2
  int lr   = lane & 15;

  v8f acc = {0.f, 0.f, 0.f, 0.f, 0.f, 0.f, 0.f, 0.f};
  const float* arow = A  + (size_t)(m0 + lr) * EDIM;
  const float* bcol = Wv + n0 + lr;

  for (int k = 0; k < DDIM; k += 4) {
    v2f a, b;
    a[0] = arow[k + 2 * half];
    a[1] = arow[k + 2 * half + 1];
    b[0] = bcol[(size_t)(k + 2 * half) * EDIM];
    b[1] = bcol[(size_t)(k + 2 * half + 1) * EDIM];
    acc = __builtin_amdgcn_wmma_f32_16x16x4_f32(false, a, false, b,
                                                (short)0, acc, false, false);
  }

#pragma unroll
  for (int vg = 0; vg < 8; ++vg) {
    int m = m0 + vg + 8 * half;     // C/D layout: VGPR j -> rows j and j+8
    int n = n0 + lr;
    float ind = segsum[m] > 0.f ? 1.f : 0.f;  // empty segment: reference adds no bv
    pooled[(size_t)m * EDIM + n] = acc[vg] + bv[n] * ind;
  }
}

// ---- kernel 5: out = pooled @ Wo + bo ------------------------------------

__global__ void k_out(const float* __restrict__ pooled, const float* __restrict__ Wo,
                      const float* __restrict__ bo, float* __restrict__ out) {
  int wave = threadIdx.x >> 5, lane = threadIdx.x & 31;
  float b0 = bo[0];
  for (int s = wave; s < NSEG; s += 8) {
    const float* pr = pooled + (size_t)s * EDIM;
    float p = 0.f;
#pragma unroll
    for (int j = 0; j < EDIM / 32; ++j)
      p = fmaf(pr[lane + 32 * j], Wo[lane + 32 * j], p);
    p = waveReduceAdd(p);
    if (lane == 0) out[s] = p + b0;
  }
}

// ---- launch --------------------------------------------------------------

extern "C" void kernel_launch(void* const* d_in, const int* in_sizes, int n_in,
                              void* d_out, int out_size, void* d_ws, size_t ws_size,
                              hipStream_t stream) {
  const float* x   = (const float*)d_in[0];
  const int*   seg = (const int*)  d_in[1];
  const float* Wk  = (const float*)d_in[2];
  const float* bk  = (const float*)d_in[3];
  const float* Wv  = (const float*)d_in[4];
  const float* bv  = (const float*)d_in[5];
  const float* Wo  = (const float*)d_in[6];
  const float* bo  = (const float*)d_in[7];
  float* out = (float*)d_out;
  int N = in_sizes[1];

  float* ws     = (float*)d_ws;
  float* logits = ws;                    // N
  float* segmax = ws + N;                // 128
  float* segsum = segmax + NSEG;         // 128
  float* wksum  = segsum + NSEG;         // 256
  float* bksum  = wksum + DDIM;          // 1 (padded)
  float* A      = ws + N + 1024;         // 128*256
  float* pooled = A + NSEG * EDIM;       // 128*256

  k_prep<<<1, 256, 0, stream>>>(Wk, bk, wksum, bksum, segmax, segsum, A);
  k_logits<<<(N + 63) / 64, 256, 0, stream>>>(x, seg, wksum, bksum, logits, segmax, N);
  k_segsum<<<(N + 4095) / 4096, 256, 0, stream>>>(logits, seg, segmax, segsum, N);
  k_accum<<<(N + R3 - 1) / R3, 256, 2 * TROWS * EDIM * sizeof(float), stream>>>(
      x, seg, logits, segmax, segsum, A, N);
  k_wmma_pooled<<<16, 256, 0, stream>>>(A, Wv, bv, segsum, pooled);
  k_out<<<1, 256, 0, stream>>>(pooled, Wo, bo, out);
}